// TrainableMMSEPerUserModule_43095701848145
// MI455X (gfx1250) — compile-verified
//
#include <hip/hip_runtime.h>
#include <hip/hip_bf16.h>
#include <math.h>

typedef __attribute__((ext_vector_type(2))) float v2f;
typedef __attribute__((ext_vector_type(8))) float v8f;

// ---------------------------------------------------------------------------
// GEMV: y[row] = (relu?)(W[row,:] . x + b[row]);  one wave32 per row.
// Rows are contiguous fp32, lanes stride in 16B chunks -> global_load_b128,
// fully coalesced 512B per wave per iteration. Pure bandwidth streaming.
// ---------------------------------------------------------------------------
__global__ __launch_bounds__(256) void gemv_kernel(const float* __restrict__ W,
                                                   const float* __restrict__ b,
                                                   const float* __restrict__ x,
                                                   float* __restrict__ y,
                                                   int rows, int cols, int do_relu)
{
    int wave = (int)((blockIdx.x * blockDim.x + threadIdx.x) >> 5);
    int lane = (int)(threadIdx.x & 31u);
    if (wave >= rows) return;

    const float* row = W + (size_t)wave * (size_t)cols;
    float sum = 0.0f;
    // cols is always a multiple of 128 (2048 or 8192)
    for (int c = lane * 4; c < cols; c += 32 * 4) {
        float4 w4 = *reinterpret_cast<const float4*>(row + c);
        float4 x4 = *reinterpret_cast<const float4*>(x + c);
        sum = fmaf(w4.x, x4.x, sum);
        sum = fmaf(w4.y, x4.y, sum);
        sum = fmaf(w4.z, x4.z, sum);
        sum = fmaf(w4.w, x4.w, sum);
    }
    // wave32 butterfly reduction
    for (int off = 16; off > 0; off >>= 1)
        sum += __shfl_down(sum, off, 32);

    if (lane == 0) {
        float v = sum + b[wave];
        if (do_relu) v = v > 0.0f ? v : 0.0f;
        y[wave] = v;
    }
}

// ---------------------------------------------------------------------------
// Build complex lower-triangular L (softplus diag) from 144 params, compute
// L @ L^H as a 16x16-padded matmul with V_WMMA_F32_16X16X4_F32, K=12 in 3
// accumulate steps. One wave32 block per MLP (blockIdx.x = 0:C, 1:R).
//
// Fragment trick: with the ISA's 16x4 A layout and 4x16 B layout, the A
// fragment of X equals the B fragment of X^T, so each matrix needs one
// fragment per K-step.
//   real = Lr.Lr^T + Li.Li^T ; imag = Li.Lr^T - Lr.Li^T  (two accumulators)
// ---------------------------------------------------------------------------
__global__ __launch_bounds__(32) void chol_outer_kernel(const float* __restrict__ pC,
                                                        const float* __restrict__ pR,
                                                        float* __restrict__ out)
{
    __shared__ float Lr[16][16];
    __shared__ float Li[16][16];

    const float* p = (blockIdx.x == 0) ? pC : pR;
    float* o = out + (size_t)blockIdx.x * 288;   // 12*12 complex = 288 floats
    const int t = (int)threadIdx.x;

    // zero the padded tiles
    for (int i = t; i < 256; i += 32) {
        (&Lr[0][0])[i] = 0.0f;
        (&Li[0][0])[i] = 0.0f;
    }
    __syncthreads();

    // scatter 78 real params (tril row-major); softplus on the diagonal
    for (int k = t; k < 78; k += 32) {
        int r = 0;
        while ((r + 1) * (r + 2) / 2 <= k) ++r;
        int c = k - r * (r + 1) / 2;
        float v = p[k];
        if (r == c) v = (v > 20.0f) ? v : log1pf(expf(v));
        Lr[r][c] = v;
    }
    // scatter 66 imag params (strict tril row-major)
    for (int k = t; k < 66; k += 32) {
        int r = 1;
        while (r * (r + 1) / 2 <= k) ++r;
        int c = k - r * (r - 1) / 2;
        Li[r][c] = p[78 + k];
    }
    __syncthreads();

    const int half = t >> 4;      // 0: K pair {0,1}, 1: K pair {2,3}
    const int m    = t & 15;      // A row / B column

    v8f cre  = {};   // Lr.Lr^T + Li.Li^T
    v8f cim1 = {};   // Li.Lr^T
    v8f cim2 = {};   // Lr.Li^T
    for (int kk = 0; kk < 3; ++kk) {
        int k0 = 4 * kk + half * 2;
        v2f ar = { Lr[m][k0], Lr[m][k0 + 1] };
        v2f ai = { Li[m][k0], Li[m][k0 + 1] };
        cre  = __builtin_amdgcn_wmma_f32_16x16x4_f32(false, ar, false, ar, (short)0, cre,  false, false);
        cre  = __builtin_amdgcn_wmma_f32_16x16x4_f32(false, ai, false, ai, (short)0, cre,  false, false);
        cim1 = __builtin_amdgcn_wmma_f32_16x16x4_f32(false, ai, false, ar, (short)0, cim1, false, false);
        cim2 = __builtin_amdgcn_wmma_f32_16x16x4_f32(false, ar, false, ai, (short)0, cim2, false, false);
    }

    // C/D layout: VGPR j, lanes 0-15 -> (M=j, N=lane); lanes 16-31 -> (M=j+8)
    for (int j = 0; j < 8; ++j) {
        int mm = j + half * 8;
        if (mm < 12 && m < 12) {
            int idx = mm * 12 + m;
            o[2 * idx]     = cre[j];
            o[2 * idx + 1] = cim1[j] - cim2[j];
        }
    }
}

// ---------------------------------------------------------------------------
// Launch: 6 GEMVs (stream-serialized to respect layer deps) + WMMA epilogue.
// Workspace (floats): h1c[2048] h1r[2048] h2c[2048] h2r[2048] pc[144] pr[144]
// ---------------------------------------------------------------------------
extern "C" void kernel_launch(void* const* d_in, const int* in_sizes, int n_in,
                              void* d_out, int out_size, void* d_ws, size_t ws_size,
                              hipStream_t stream)
{
    (void)in_sizes; (void)n_in; (void)out_size; (void)ws_size;

    const float* x   = (const float*)d_in[0];
    const float* Wc1 = (const float*)d_in[3];
    const float* bc1 = (const float*)d_in[4];
    const float* Wc2 = (const float*)d_in[5];
    const float* bc2 = (const float*)d_in[6];
    const float* Wc3 = (const float*)d_in[7];
    const float* bc3 = (const float*)d_in[8];
    const float* Wr1 = (const float*)d_in[9];
    const float* br1 = (const float*)d_in[10];
    const float* Wr2 = (const float*)d_in[11];
    const float* br2 = (const float*)d_in[12];
    const float* Wr3 = (const float*)d_in[13];
    const float* br3 = (const float*)d_in[14];

    float* ws  = (float*)d_ws;
    float* h1c = ws;
    float* h1r = ws + 2048;
    float* h2c = ws + 4096;
    float* h2r = ws + 6144;
    float* pc  = ws + 8192;
    float* pr  = ws + 8192 + 144;

    const dim3 blk(256);
    // layer 1: 2048 rows x 8192 cols (independent for C and R paths)
    gemv_kernel<<<dim3(2048 / 8), blk, 0, stream>>>(Wc1, bc1, x, h1c, 2048, 8192, 1);
    gemv_kernel<<<dim3(2048 / 8), blk, 0, stream>>>(Wr1, br1, x, h1r, 2048, 8192, 1);
    // layer 2: 2048 x 2048
    gemv_kernel<<<dim3(2048 / 8), blk, 0, stream>>>(Wc2, bc2, h1c, h2c, 2048, 2048, 1);
    gemv_kernel<<<dim3(2048 / 8), blk, 0, stream>>>(Wr2, br2, h1r, h2r, 2048, 2048, 1);
    // layer 3: 144 x 2048, no relu
    gemv_kernel<<<dim3(18), blk, 0, stream>>>(Wc3, bc3, h2c, pc, 144, 2048, 0);
    gemv_kernel<<<dim3(18), blk, 0, stream>>>(Wr3, br3, h2r, pr, 144, 2048, 0);
    // complex L L^H via WMMA, one wave per MLP
    chol_outer_kernel<<<dim3(2), dim3(32), 0, stream>>>(pc, pr, (float*)d_out);
}